// S4D_rnn_6743098655677
// MI455X (gfx1250) — compile-verified
//
#include <hip/hip_runtime.h>

// ---------------------------------------------------------------------------
// S4D diagonal SSM scan for MI455X (gfx1250, wave32).
//
// Roofline: output all_xs = 268MB written once -> HBM-write bound (~11.5us at
// 23.3TB/s). Strategy: one wave per (b,d) channel, lane = state index n<32
// (conjugate symmetry gives the upper 32 states for free), non-temporal
// streaming stores, async global->LDS staging of the u row, and the C.x
// einsum done with V_WMMA_F32_16X16X4_F32 on 16-timestep tiles.
// ---------------------------------------------------------------------------

typedef __attribute__((ext_vector_type(2))) float v2f;
typedef __attribute__((ext_vector_type(8))) float v8f;
typedef int v4i __attribute__((vector_size(16)));   // matches builtin's V4i

#define AS1 __attribute__((address_space(1)))
#define AS3 __attribute__((address_space(3)))

static constexpr int kD  = 128;   // channels
static constexpr int kN2 = 32;    // states kept (of 64; upper half = conj)
static constexpr int kB  = 4;     // batch
static constexpr int kL  = 1024;  // sequence length
static constexpr int kTS = 33;    // padded LDS row stride for the 16x32 tile

__global__ __launch_bounds__(32) void s4d_scan_wmma(
    const float* __restrict__ us,          // (B, D, L)
    const float* __restrict__ log_dt,      // (D,)
    const float* __restrict__ log_A_real,  // (D, 32)
    const float* __restrict__ A_imag,      // (D, 32)
    const float* __restrict__ B_re,        // (D, 64)
    const float* __restrict__ B_im,        // (D, 64)
    const float* __restrict__ C_re,        // (D, 32)
    const float* __restrict__ C_im,        // (D, 32) (unused: Re(y) only needs C_re)
    float* __restrict__ out)               // [all_xs B*L*D*64 cplx][ys B*D*L f32]
{
  __shared__ __align__(16) float su[kL];        // staged u row (4KB, wave-private)
  __shared__ __align__(16) float tw[16 * kTS];  // Re(x) tile: 16 t-rows x 32 states

  const int lane = threadIdx.x;        // 0..31, one wave per block
  const int pair = blockIdx.x;         // pair = b*128 + d
  const int b = pair >> 7;
  const int d = pair & (kD - 1);

  const float* urow = us + (size_t)pair * kL;   // us[(b*D+d)*L + t]

  // ---- stage u row into LDS with CDNA5 async copies (8 x b128 = 4KB) ------
#if __has_builtin(__builtin_amdgcn_global_load_async_to_lds_b128)
#pragma unroll
  for (int c = 0; c < 8; ++c) {
    __builtin_amdgcn_global_load_async_to_lds_b128(
        (AS1 v4i*)(uintptr_t)(urow + c * 128 + lane * 4),
        (AS3 v4i*)(uintptr_t)(su + c * 128 + lane * 4),
        0, 0);
  }
#else
#pragma unroll
  for (int c = 0; c < 8; ++c)
    ((float4*)su)[c * 32 + lane] = ((const float4*)urow)[c * 32 + lane];
#endif

  // ---- per-lane discretization (overlapped with the async copies) ---------
  const float dt   = __expf(log_dt[d]);
  const float Are  = -__expf(log_A_real[d * kN2 + lane]);
  const float Aim  = A_imag[d * kN2 + lane];
  const float dtAr = dt * Are, dtAi = dt * Aim;
  const float den_r = 1.0f - 0.5f * dtAr, den_i = -0.5f * dtAi;
  const float num_r = 1.0f + 0.5f * dtAr, num_i =  0.5f * dtAi;
  const float inv   = 1.0f / (den_r * den_r + den_i * den_i);
  const float dAre  = (num_r * den_r + num_i * den_i) * inv;
  const float dAim  = (num_i * den_r - num_r * den_i) * inv;
  const float tbr   = dt * B_re[d * 64 + lane];
  const float tbi   = dt * B_im[d * 64 + lane];
  const float dBre  = (tbr * den_r + tbi * den_i) * inv;
  const float dBim  = (tbi * den_r - tbr * den_i) * inv;
  (void)C_im;  // Re(y) = 2*sum C_re*Re(x) by conjugate symmetry

  // ---- loop-invariant WMMA B tiles: B[k][j] = 2*C_re[k] broadcast over j ---
  const int khalf = (lane >> 4) << 1;  // vgpr j, lane-half h hold row K = j + 2h
  const float* crow = C_re + d * kN2;
  v2f bt[8];
#pragma unroll
  for (int kt = 0; kt < 8; ++kt) {
    const int k = kt * 4 + khalf;
    bt[kt].x = 2.0f * crow[k];
    bt[kt].y = 2.0f * crow[k + 1];
  }

#if __has_builtin(__builtin_amdgcn_global_load_async_to_lds_b128)
#if __has_builtin(__builtin_amdgcn_s_wait_asynccnt)
  __builtin_amdgcn_s_wait_asynccnt(0);
#else
  asm volatile("s_wait_asynccnt 0" ::: "memory");
#endif
#endif
  asm volatile("" ::: "memory");  // keep LDS reads after the async wait

  // ---- output pointers -----------------------------------------------------
  v2f* xs = (v2f*)out + ((size_t)b * kL * kD + (size_t)d) * 64 + lane;
  float* ysr = out + (size_t)kB * kL * kD * 64 * 2 + (size_t)pair * kL;

  const int yoff = (lane & 7) + ((lane >> 4) << 3);  // which y[t] this lane holds
  const int arow = (lane & 15) * kTS + khalf;        // WMMA A-operand LDS row

  float xr = 0.0f, xi = 0.0f;

  for (int t0 = 0; t0 < kL; t0 += 16) {
    // -------- 16 recurrence steps, streaming NT stores of x and conj(x) ----
#pragma unroll
    for (int k = 0; k < 16; ++k) {
      const float u  = su[t0 + k];  // LDS broadcast read
      const float nr = fmaf(dAre, xr, fmaf(-dAim, xi, dBre * u));
      const float ni = fmaf(dAre, xi, fmaf( dAim, xr, dBim * u));
      xr = nr; xi = ni;

      v2f v;  v.x = xr;  v.y = xi;
      __builtin_nontemporal_store(v, xs);          // x[n]
      v2f vc; vc.x = xr; vc.y = -xi;
      __builtin_nontemporal_store(vc, xs + 32);    // x[n+32] = conj(x[n])
      xs += kD * 64;                               // advance one timestep

      tw[k * kTS + lane] = xr;                     // Re(x) tile for the einsum
    }

    // -------- y[t0..t0+15] = Xre(16x32) * (2*C_re) via 8 chained WMMAs -----
    v8f acc = {0.f, 0.f, 0.f, 0.f, 0.f, 0.f, 0.f, 0.f};
#pragma unroll
    for (int kt = 0; kt < 8; ++kt) {
      v2f a;
      a.x = tw[arow + kt * 4];
      a.y = tw[arow + kt * 4 + 1];
      acc = __builtin_amdgcn_wmma_f32_16x16x4_f32(
          false, a, false, bt[kt], (short)0, acc, false, false);
    }
    // D[M,N] = y[M] for every N (B broadcast). vgpr r holds M = r + 8*(lane>>4).
    const bool b0 = (lane & 1), b1 = (lane & 2), b2 = (lane & 4);
    const float s0 = b0 ? acc[1] : acc[0];
    const float s1 = b0 ? acc[3] : acc[2];
    const float s2 = b0 ? acc[5] : acc[4];
    const float s3 = b0 ? acc[7] : acc[6];
    const float u0 = b1 ? s1 : s0;
    const float u1 = b1 ? s3 : s2;
    const float y  = b2 ? u1 : u0;
    ysr[t0 + yoff] = y;   // lanes l and l^8 write identical value: deterministic
  }
}

extern "C" void kernel_launch(void* const* d_in, const int* in_sizes, int n_in,
                              void* d_out, int out_size, void* d_ws, size_t ws_size,
                              hipStream_t stream) {
  (void)in_sizes; (void)n_in; (void)out_size; (void)d_ws; (void)ws_size;
  s4d_scan_wmma<<<dim3(kB * kD), dim3(32), 0, stream>>>(
      (const float*)d_in[0],  // us
      (const float*)d_in[1],  // log_dt
      (const float*)d_in[2],  // log_A_real
      (const float*)d_in[3],  // A_imag
      (const float*)d_in[4],  // B_re
      (const float*)d_in[5],  // B_im
      (const float*)d_in[6],  // C_re
      (const float*)d_in[7],  // C_im
      (float*)d_out);
}